// BNESNN_46213848105973
// MI455X (gfx1250) — compile-verified
//
#include <hip/hip_runtime.h>

// CDNA5 / gfx1250, wave32. Fused BNESNN forward + STDP outer-product backward.
// V_WMMA_F32_16X16X4_F32: A=16x4 f32 (2 VGPR/lane), B=4x16 f32 (2 VGPR/lane),
// C/D=16x16 f32 (8 VGPR/lane).  Layouts per CDNA5 ISA 7.12.2:
//   A: lanes 0-15 hold M=lane, VGPR0=K+0, VGPR1=K+1; lanes 16-31 hold K+2,K+3
//   B: VGPR0 lanes 0-15 = row K+0 (N=lane), lanes 16-31 = row K+2; VGPR1 = K+1/K+3
//   C/D: VGPR j, lanes 0-15 -> (M=j, N=lane); lanes 16-31 -> (M=j+8, N=lane-16)

typedef __attribute__((ext_vector_type(2))) float v2f;
typedef __attribute__((ext_vector_type(8))) float v8f;

static __device__ __forceinline__ v8f wmma_f32(v2f a, v2f b, v8f c) {
    return __builtin_amdgcn_wmma_f32_16x16x4_f32(
        /*neg_a=*/false, a, /*neg_b=*/false, b,
        /*c_mod=*/(short)0, c, /*reuse_a=*/false, /*reuse_b=*/false);
}

#define WAVES_PER_BLK 8          // blockDim.x == 256 -> 8 wave32
#define DW_ELEMS      864        // 12*24 + 24*24

__global__ __launch_bounds__(256)
void bnesnn_main(const float* __restrict__ x1,   // [B,12]
                 const float* __restrict__ x2,   // [B,24]
                 const float* __restrict__ W0,   // [12,12]
                 const float* __restrict__ W1,   // [12,24]
                 const float* __restrict__ W2,   // [24,24]
                 float* __restrict__ partials,   // [gridDim.x, 864]
                 float* __restrict__ s0_out,     // [B,12]
                 float* __restrict__ s1_out,     // [B,24]
                 int B)
{
    const int lane   = threadIdx.x & 31;
    const int waveIb = threadIdx.x >> 5;
    const int waveId = blockIdx.x * WAVES_PER_BLK + waveIb;
    const int nWaves = gridDim.x * WAVES_PER_BLK;

    const int n    = lane & 15;   // N index in B/C/D tiles; M index for A rows
    const int half = lane >> 4;   // lane half selects K+0/K+1 vs K+2/K+3
    const int koff = half * 2;

    // ---------------- weights in B-matrix layout (wave-invariant) -------------
    v2f bW0[3];                       // W0: K=12 -> 3 K-steps, N pad 12->16
    #pragma unroll
    for (int ks = 0; ks < 3; ++ks) {
        int k = ks * 4 + koff;
        bW0[ks].x = (n < 12) ? W0[k * 12 + n]       : 0.0f;
        bW0[ks].y = (n < 12) ? W0[(k + 1) * 12 + n] : 0.0f;
    }
    v2f bW1[2][3], bW2[2][6];         // N=24 -> 2 N-tiles
    #pragma unroll
    for (int t = 0; t < 2; ++t) {
        int  col = t * 16 + n;
        bool cv  = (col < 24);
        #pragma unroll
        for (int ks = 0; ks < 3; ++ks) {
            int k = ks * 4 + koff;
            bW1[t][ks].x = cv ? W1[k * 24 + col]       : 0.0f;
            bW1[t][ks].y = cv ? W1[(k + 1) * 24 + col] : 0.0f;
        }
        #pragma unroll
        for (int ks = 0; ks < 6; ++ks) {
            int k = ks * 4 + koff;
            bW2[t][ks].x = cv ? W2[k * 24 + col]       : 0.0f;
            bW2[t][ks].y = cv ? W2[(k + 1) * 24 + col] : 0.0f;
        }
    }

    // ---------------- backward accumulators (carried over all chunks) ---------
    const v8f zero8 = {};
    v8f cDW1[2];                      // dw1: 1 M-tile x 2 N-tiles
    v8f cDW2[2][2];                   // dw2: 2 M-tiles x 2 N-tiles
    cDW1[0] = zero8; cDW1[1] = zero8;
    #pragma unroll
    for (int mt = 0; mt < 2; ++mt) { cDW2[mt][0] = zero8; cDW2[mt][1] = zero8; }

    const int nChunks = B >> 4;       // B is a multiple of 16 (B = 1<<20)
    for (int ch = waveId; ch < nChunks; ch += nWaves) {
        const int base = ch << 4;
        const int rowA = base + n;    // this lane's A-matrix row (M = lane&15)

        // prefetch next chunk's streams (global_prefetch_b8)
        int nxt = ch + nWaves;
        if (nxt < nChunks) {
            __builtin_prefetch(&x1[(size_t)(nxt << 4) * 12], 0, 0);
            __builtin_prefetch(&x2[(size_t)(nxt << 4) * 24], 0, 0);
        }

        // ---- layer 1: i0 = x1 @ W0 ; s0 = (i0 > 0.5) ------------------------
        v8f c0 = zero8;
        #pragma unroll
        for (int ks = 0; ks < 3; ++ks) {
            int k = ks * 4 + koff;
            v2f a;
            a.x = x1[rowA * 12 + k];
            a.y = x1[rowA * 12 + k + 1];
            c0 = wmma_f32(a, bW0[ks], c0);
        }
        #pragma unroll
        for (int j = 0; j < 8; ++j) {
            float sp = (c0[j] > 0.5f) ? 1.0f : 0.0f;
            if (n < 12) s0_out[(base + j + 8 * half) * 12 + n] = sp;
        }

        // ---- layer 2: i1 = s0 @ W1 + x2 @ W2 ; s1 = (i1 > 0.5) --------------
        // A operands; s0 reloaded in A layout (same-wave store->load: ordered, L2 hot)
        v2f aS0[3], aX2[6];
        #pragma unroll
        for (int ks = 0; ks < 3; ++ks) {
            int k = ks * 4 + koff;
            aS0[ks].x = s0_out[rowA * 12 + k];
            aS0[ks].y = s0_out[rowA * 12 + k + 1];
        }
        #pragma unroll
        for (int ks = 0; ks < 6; ++ks) {
            int k = ks * 4 + koff;
            aX2[ks].x = x2[rowA * 24 + k];
            aX2[ks].y = x2[rowA * 24 + k + 1];
        }
        #pragma unroll
        for (int t = 0; t < 2; ++t) {
            v8f c1 = zero8;
            #pragma unroll
            for (int ks = 0; ks < 3; ++ks) c1 = wmma_f32(aS0[ks], bW1[t][ks], c1);
            #pragma unroll
            for (int ks = 0; ks < 6; ++ks) c1 = wmma_f32(aX2[ks], bW2[t][ks], c1);
            int col = t * 16 + n;
            #pragma unroll
            for (int j = 0; j < 8; ++j) {
                float sp = (c1[j] > 0.5f) ? 1.0f : 0.0f;
                if (col < 24) s1_out[(base + j + 8 * half) * 24 + col] = sp;
            }
        }

        // ---- backward: dw1 += s0^T s1 ; dw2 += x2^T s1  (K = 16 rows/chunk) --
        #pragma unroll
        for (int g = 0; g < 4; ++g) {
            int b0 = base + g * 4 + koff;      // this lane's two K rows
            v2f bS1[2];
            #pragma unroll
            for (int t = 0; t < 2; ++t) {
                int  col = t * 16 + n;
                bool cv  = (col < 24);
                bS1[t].x = cv ? s1_out[b0 * 24 + col]       : 0.0f;
                bS1[t].y = cv ? s1_out[(b0 + 1) * 24 + col] : 0.0f;
            }
            v2f aT1;                            // A = s0^T (M=12 pad 16)
            aT1.x = (n < 12) ? s0_out[b0 * 12 + n]       : 0.0f;
            aT1.y = (n < 12) ? s0_out[(b0 + 1) * 12 + n] : 0.0f;
            cDW1[0] = wmma_f32(aT1, bS1[0], cDW1[0]);
            cDW1[1] = wmma_f32(aT1, bS1[1], cDW1[1]);
            #pragma unroll
            for (int mt = 0; mt < 2; ++mt) {    // A = x2^T (M=24 -> 2 M-tiles)
                int m = mt * 16 + n;
                v2f aT2;
                aT2.x = (m < 24) ? x2[b0 * 24 + m]       : 0.0f;
                aT2.y = (m < 24) ? x2[(b0 + 1) * 24 + m] : 0.0f;
                cDW2[mt][0] = wmma_f32(aT2, bS1[0], cDW2[mt][0]);
                cDW2[mt][1] = wmma_f32(aT2, bS1[1], cDW2[mt][1]);
            }
        }
    }

    // ---------------- deterministic block reduction of dw accumulators -------
    __shared__ float red[WAVES_PER_BLK * DW_ELEMS];
    float* slice = red + waveIb * DW_ELEMS;     // each slot written exactly once
    #pragma unroll
    for (int j = 0; j < 8; ++j) {
        int M = j + 8 * half;
        #pragma unroll
        for (int t = 0; t < 2; ++t) {
            int col = t * 16 + n;
            if (col < 24) {
                if (M < 12) slice[M * 24 + col] = cDW1[t][j];
                #pragma unroll
                for (int mt = 0; mt < 2; ++mt) {
                    int Mr = mt * 16 + M;
                    if (Mr < 24) slice[288 + Mr * 24 + col] = cDW2[mt][t][j];
                }
            }
        }
    }
    __syncthreads();
    for (int idx = threadIdx.x; idx < DW_ELEMS; idx += blockDim.x) {
        float s = 0.0f;
        #pragma unroll
        for (int w = 0; w < WAVES_PER_BLK; ++w) s += red[w * DW_ELEMS + idx];
        partials[(size_t)blockIdx.x * DW_ELEMS + idx] = s;
    }
}

// Fixed-order final reduction: bitwise deterministic (no float atomics).
__global__ __launch_bounds__(1024)
void bnesnn_reduce(const float* __restrict__ partials,
                   float* __restrict__ dw_out,   // 864 contiguous floats
                   int nBlocks)
{
    int idx = threadIdx.x;
    if (idx < DW_ELEMS) {
        float s = 0.0f;
        for (int b = 0; b < nBlocks; ++b) s += partials[(size_t)b * DW_ELEMS + idx];
        dw_out[idx] = s;
    }
}

extern "C" void kernel_launch(void* const* d_in, const int* in_sizes, int n_in,
                              void* d_out, int out_size, void* d_ws, size_t ws_size,
                              hipStream_t stream) {
    const float* x1 = (const float*)d_in[0];
    const float* x2 = (const float*)d_in[1];
    const float* W0 = (const float*)d_in[2];
    const float* W1 = (const float*)d_in[3];
    const float* W2 = (const float*)d_in[4];
    const int B = in_sizes[0] / 12;

    float* out = (float*)d_out;
    float* dw  = out;                          // dw1[288] ++ dw2[576] contiguous
    float* s0  = out + DW_ELEMS;               // [B,12]
    float* s1  = out + DW_ELEMS + (size_t)B * 12;  // [B,24]
    float* partials = (float*)d_ws;

    // 512 blocks x 256 threads = 4096 wave32 -> 16 chunks of 16 rows per wave.
    int nBlocks = 512;
    int maxBlk  = (int)(ws_size / (DW_ELEMS * sizeof(float)));
    if (nBlocks > maxBlk) nBlocks = maxBlk;
    if (nBlocks < 1) nBlocks = 1;

    bnesnn_main<<<nBlocks, 256, 0, stream>>>(x1, x2, W0, W1, W2,
                                             partials, s0, s1, B);
    bnesnn_reduce<<<1, 1024, 0, stream>>>(partials, dw, nBlocks);
}